// AlgorithmExecutor_49821620633837
// MI455X (gfx1250) — compile-verified
//
#include <hip/hip_runtime.h>
#include <hip/hip_bf16.h>
#include <math.h>

typedef __attribute__((ext_vector_type(2))) float v2f;
typedef __attribute__((ext_vector_type(8))) float v8f;

#define HD   128      // hidden dim
#define MT   4        // M-tiles (of 16 rows) per block
#define ROWS (MT*16)  // 64 rows per block
#define STR  260      // padded LDS row stride (260%64==4 -> conflict-free, 8B aligned)

// ---------------- encoder: h[n,c] = x[n]*W_enc[c] + b_enc[c] ----------------
__global__ void k_encode(const float* __restrict__ x, const float* __restrict__ Wenc,
                         const float* __restrict__ benc, float* __restrict__ h, int n) {
    int i = blockIdx.x * blockDim.x + threadIdx.x;
    if (i < n * HD) {
        int node = i >> 7, c = i & (HD - 1);
        h[i] = fmaf(x[node], Wenc[c], benc[c]);
    }
}

// ------- message GEMM (E x 257) @ (257 x 128), relu, atomic-max scatter -------
// Block = 256 threads (8 waves). Each block: 64 edges (4 M-tiles). Each wave: 16 cols.
// B fragments are loaded once per K-step and reused by 4 WMMAs (register M-blocking).
__global__ __launch_bounds__(256) void k_message(
        const float* __restrict__ h, const int* __restrict__ ei,
        const float* __restrict__ ea, const float* __restrict__ Wmsg,
        const float* __restrict__ bmsg, unsigned int* __restrict__ aggr, int E) {
    __shared__ float sA[ROWS * STR];   // combined rows [64][256]
    __shared__ int   sSrc[ROWS];
    __shared__ int   sTgt[ROWS];
    __shared__ float sAttr[ROWS];

    const int tid   = threadIdx.x;
    const int ebase = blockIdx.x * ROWS;

    if (tid < ROWS) {
        int e = ebase + tid;
        if (e >= E) e = E - 1;               // pad by duplication; atomic max is idempotent
        sSrc[tid]  = ei[e];                  // edge_index[0] = source j
        sTgt[tid]  = ei[E + e];              // edge_index[1] = target i
        sAttr[tid] = ea[e];
    }
    __syncthreads();

    // stage combined = [h[tgt] | h[src]] for 64 edges (64 x 256 floats)
    for (int i = tid; i < ROWS * 256; i += 256) {
        int r = i >> 8, k = i & 255;
        int node = (k < HD) ? sTgt[r] : sSrc[r];
        sA[r * STR + k] = h[(size_t)node * HD + (k & (HD - 1))];
    }
    __syncthreads();

    const int wave = tid >> 5;        // 0..7 -> 16-col tile
    const int lane = tid & 31;
    const int half = lane >> 4;       // K sub-pair selector per f32 WMMA layout
    const int idx  = lane & 15;       // A: row-in-tile; B/D: col-in-tile
    const int col  = wave * 16 + idx; // output column 0..127

    const float* Bcol = Wmsg + col;   // W_msg[k][col] = Wmsg[k*128 + col]
    v8f acc[MT] = {};

    for (int k = 0; k < 256; k += 4) {
        int kp = k + half * 2;
        v2f b;
        b.x = Bcol[(size_t)kp * HD];
        b.y = Bcol[(size_t)(kp + 1) * HD];
#pragma unroll
        for (int t = 0; t < MT; ++t) {
            v2f a = *(const v2f*)&sA[(t * 16 + idx) * STR + kp];
            acc[t] = __builtin_amdgcn_wmma_f32_16x16x4_f32(
                false, a, false, b, (short)0, acc[t], false, false);
        }
    }

    // K=256 term (edge_attr column of W_msg), bias, relu, atomic-max scatter.
    const float wlast = Wmsg[(size_t)256 * HD + col];
    const float bias  = bmsg[col];
#pragma unroll
    for (int t = 0; t < MT; ++t) {
#pragma unroll
        for (int r = 0; r < 8; ++r) {
            int erow = t * 16 + r + half * 8;      // D layout: lanes>=16 hold M=r+8
            float v  = acc[t][r] + sAttr[erow] * wlast + bias;
            v = v > 0.0f ? v : 0.0f;               // relu => v >= 0
            // non-negative floats: uint ordering == float ordering
            atomicMax(aggr + (size_t)sTgt[erow] * HD + col, __float_as_uint(v));
        }
    }
}

// ------------- update GEMM: relu([h|aggr](N x 256) @ W_upd(256x128)) -------------
__global__ __launch_bounds__(256) void k_update(
        const float* __restrict__ h, const float* __restrict__ aggr,
        const float* __restrict__ Wupd, const float* __restrict__ bupd,
        float* __restrict__ h2, int n) {
    __shared__ float sA[ROWS * STR];

    const int tid   = threadIdx.x;
    const int nbase = blockIdx.x * ROWS;

    for (int i = tid; i < ROWS * 256; i += 256) {
        int r = i >> 8, k = i & 255;
        int node = nbase + r;
        if (node >= n) node = n - 1;               // clamp; tail stores are guarded below
        const float* buf = (k < HD) ? h : aggr;
        sA[r * STR + k] = buf[(size_t)node * HD + (k & (HD - 1))];
    }
    __syncthreads();

    const int wave = tid >> 5;
    const int lane = tid & 31;
    const int half = lane >> 4;
    const int idx  = lane & 15;
    const int col  = wave * 16 + idx;

    const float* Bcol = Wupd + col;
    v8f acc[MT] = {};

    for (int k = 0; k < 256; k += 4) {
        int kp = k + half * 2;
        v2f b;
        b.x = Bcol[(size_t)kp * HD];
        b.y = Bcol[(size_t)(kp + 1) * HD];
#pragma unroll
        for (int t = 0; t < MT; ++t) {
            v2f a = *(const v2f*)&sA[(t * 16 + idx) * STR + kp];
            acc[t] = __builtin_amdgcn_wmma_f32_16x16x4_f32(
                false, a, false, b, (short)0, acc[t], false, false);
        }
    }

    const float bias = bupd[col];
#pragma unroll
    for (int t = 0; t < MT; ++t) {
#pragma unroll
        for (int r = 0; r < 8; ++r) {
            int nrow = nbase + t * 16 + r + half * 8;
            float v  = acc[t][r] + bias;
            v = v > 0.0f ? v : 0.0f;
            if (nrow < n) h2[(size_t)nrow * HD + col] = v;
        }
    }
}

// ------------- decode: out[n] = sigmoid(h2[n,:] . W_dec + b_dec) -------------
__global__ void k_decode(const float* __restrict__ h2, const float* __restrict__ Wdec,
                         const float* __restrict__ bdec, float* __restrict__ out, int n) {
    int node = blockIdx.x * (blockDim.x >> 5) + (threadIdx.x >> 5);
    int lane = threadIdx.x & 31;
    if (node >= n) return;
    float s = 0.0f;
    for (int k = lane; k < HD; k += 32)
        s = fmaf(h2[(size_t)node * HD + k], Wdec[k], s);
    for (int off = 16; off > 0; off >>= 1)
        s += __shfl_down(s, off, 32);
    if (lane == 0)
        out[node] = 1.0f / (1.0f + expf(-(s + bdec[0])));
}

// ------------- column sums of h2 for the termination head -------------
__global__ void k_colsum(const float* __restrict__ h2, float* __restrict__ colsum, int n) {
    int c = threadIdx.x;   // 128 threads
    float s = 0.0f;
    for (int node = blockIdx.x; node < n; node += gridDim.x)
        s += h2[(size_t)node * HD + c];
    atomicAdd(colsum + c, s);
}

// ------------- term = sigmoid(mean(h2) . W_term + b_term) -------------
__global__ void k_term(const float* __restrict__ colsum, const float* __restrict__ Wterm,
                       const float* __restrict__ bterm, float* __restrict__ out_term, int n) {
    __shared__ float red[HD];
    int t = threadIdx.x;
    red[t] = (colsum[t] / (float)n) * Wterm[t];
    __syncthreads();
    for (int s = HD / 2; s > 0; s >>= 1) {
        if (t < s) red[t] += red[t + s];
        __syncthreads();
    }
    if (t == 0) *out_term = 1.0f / (1.0f + expf(-(red[0] + bterm[0])));
}

extern "C" void kernel_launch(void* const* d_in, const int* in_sizes, int n_in,
                              void* d_out, int out_size, void* d_ws, size_t ws_size,
                              hipStream_t stream) {
    const float* x     = (const float*)d_in[0];
    const int*   ei    = (const int*)d_in[1];
    const float* ea    = (const float*)d_in[2];
    const float* Wenc  = (const float*)d_in[3];
    const float* benc  = (const float*)d_in[4];
    const float* Wmsg  = (const float*)d_in[5];
    const float* bmsg  = (const float*)d_in[6];
    const float* Wupd  = (const float*)d_in[7];
    const float* bupd  = (const float*)d_in[8];
    const float* Wdec  = (const float*)d_in[9];
    const float* bdec  = (const float*)d_in[10];
    const float* Wterm = (const float*)d_in[11];
    const float* bterm = (const float*)d_in[12];

    const int N = in_sizes[0];       // 50000
    const int E = in_sizes[2];       // 800000

    float* h      = (float*)d_ws;
    float* aggr   = h    + (size_t)N * HD;
    float* h2     = aggr + (size_t)N * HD;
    float* colsum = h2   + (size_t)N * HD;
    float* out    = (float*)d_out;

    hipMemsetAsync(aggr, 0, (size_t)N * HD * sizeof(float), stream);   // relu>=0 => 0-init == -inf fill
    hipMemsetAsync(colsum, 0, HD * sizeof(float), stream);

    k_encode<<<(N * HD + 255) / 256, 256, 0, stream>>>(x, Wenc, benc, h, N);
    k_message<<<(E + ROWS - 1) / ROWS, 256, 0, stream>>>(h, ei, ea, Wmsg, bmsg,
                                                         (unsigned int*)aggr, E);
    k_update<<<(N + ROWS - 1) / ROWS, 256, 0, stream>>>(h, aggr, Wupd, bupd, h2, N);
    k_decode<<<(N + 7) / 8, 256, 0, stream>>>(h2, Wdec, bdec, out, N);
    k_colsum<<<240, HD, 0, stream>>>(h2, colsum, N);
    k_term<<<1, HD, 0, stream>>>(colsum, Wterm, bterm, out + N, N);
}